// Dynamicor_12945031430867
// MI455X (gfx1250) — compile-verified
//
#include <hip/hip_runtime.h>
#include <math.h>

// ---------------------------------------------------------------------------
// MI455X (gfx1250, wave32) implementation.
//
// Per case c (2048 cases): nodes [c*8192, c*8192+336) are "idx0",
// [c*8192+336, c*8192+672) are "idx1".  All math in f32 (memory-bound:
// ~47 MB useful traffic @ 23.3 TB/s ~ 2us; AI ~ 0.6 flop/byte).
//
// One workgroup of 352 threads (11 wave32 waves) per case.
//   stage  : n0 coords -> LDS via GLOBAL_LOAD_ASYNC_TO_LDS_B64 (ASYNCcnt)
//   phase 1: per-node tau -> LDS (roll(-1) coupling), n0/n0[i+1] from LDS
//   phase 2: per-node force partials
//   reduce : shfl_xor within wave, then V_WMMA_F32_16X16X4_F32 across waves
//            (M = quantity, K = wave index, B = ones -> row sums in f32).
// ---------------------------------------------------------------------------

typedef __attribute__((ext_vector_type(2))) float v2f;
typedef __attribute__((ext_vector_type(8))) float v8f;

// Builtin prototype (from hipcc diagnostic): param 1 is
//   'int __attribute__((vector_size(8))) __device__ *'  (addrspace(1), non-const)
typedef int v2i_vs __attribute__((vector_size(8)));
typedef __attribute__((address_space(1))) v2i_vs* g_v2i_ptr;
typedef __attribute__((address_space(3))) v2i_vs* l_v2i_ptr;

#define NODES      336
#define CASE_NODES 8192
#define BLOCK      352            // 11 waves of 32
#define ALPHA_F    0.10471975511965977f   // 6*pi/180
#define ACOUSTIC_F 347.18872f             // sqrt(1.4*287*300)
#define MIU_F      1.9e-05f

#if defined(__has_builtin)
#  if __has_builtin(__builtin_amdgcn_global_load_async_to_lds_b64)
#    define HAVE_ASYNC_LDS 1
#  else
#    define HAVE_ASYNC_LDS 0
#  endif
#else
#  define HAVE_ASYNC_LDS 0
#endif

__global__ __launch_bounds__(BLOCK) void
dynamicor_case_kernel(const float* __restrict__ coords,
                      const float* __restrict__ fields,
                      const float* __restrict__ design,
                      float* __restrict__ out)
{
    __shared__ float2 s_n0[NODES];  // staged idx0 coords (read twice: i, i+1)
    __shared__ float  s_tau[NODES];
    __shared__ float  s_part[256];  // [wave 0..15][quantity 0..15], zero padded

    const int  i    = threadIdx.x;            // node index within case
    const long long base = (long long)blockIdx.x * CASE_NODES;

    const float2* cptr = (const float2*)coords;
    const float4* fptr = (const float4*)fields;

    // Early prefetch of this case's other regions (global_prefetch_b8).
    __builtin_prefetch(fields + base * 4, 0, 0);
    __builtin_prefetch(design + base * 5, 0, 0);

    if (i < 256) s_part[i] = 0.0f;

    // ---- Stage n0 coords into LDS (async DMA path if toolchain has it) ----
#if HAVE_ASYNC_LDS
    if (i < NODES) {
        float* gsrc = const_cast<float*>(coords + (base + i) * 2);
        __builtin_amdgcn_global_load_async_to_lds_b64(
            (g_v2i_ptr)gsrc,
            (l_v2i_ptr)&s_n0[i],
            0, 0);
    }
#  if __has_builtin(__builtin_amdgcn_s_wait_asynccnt)
    __builtin_amdgcn_s_wait_asynccnt(0);
#  else
    asm volatile("s_wait_asynccnt 0x0" ::: "memory");
#  endif
#else
    if (i < NODES) s_n0[i] = cptr[base + i];
#endif
    __syncthreads();

    float fx = 0.0f, fy = 0.0f, d3 = 0.0f, d4 = 0.0f;
    float Txv = 0.0f, Tyv = 0.0f, Tn = 1.0f, Px = 0.0f, Py = 0.0f, tau = 0.0f;
    int   ip1 = 0;

    if (i < NODES) {
        ip1 = (i + 1 == NODES) ? 0 : i + 1;

        float2 n0  = s_n0[i];
        float2 n0n = s_n0[ip1];
        float2 n1  = cptr[base + NODES + i];
        float4 f0  = fptr[base + i];
        float4 f1  = fptr[base + NODES + i];
        const float* dg = design + (base + i) * 5ll;
        d3 = dg[3];
        d4 = dg[4];

        // T = n0[i+1] - n0[i];  N = (T.y, -T.x);  |N| == |T|
        Txv = n0n.x - n0.x;
        Tyv = n0n.y - n0.y;
        Tn  = sqrtf(Txv * Txv + Tyv * Tyv);

        float pt  = f0.x;              // pt_ave == pt
        float Ptn = pt * Tn;
        Px = Ptn * Tyv / Tn;           // Pt_n * N.x / N_norm
        Py = Ptn * (-Txv) / Tn;        // Pt_n * N.y / N_norm

        // tau_i = MIU * du / delta
        float du    = (f1.z * Txv + f1.w * Tyv) / Tn;
        float ddx   = n1.x - n0.x;
        float ddy   = n1.y - n0.y;
        float delta = sqrtf(ddx * ddx + ddy * ddy);
        tau = MIU_F * du / delta;
        s_tau[i] = tau;
    }
    __syncthreads();

    if (i < NODES) {
        float tau_ave = 0.5f * (tau + s_tau[ip1]);   // roll(-1) coupling
        float T_n     = tau_ave * Tn;
        float Txf     = 50.0f * T_n * Txv / Tn;
        float Tyf     = 50.0f * T_n * Tyv / Tn;
        fx = -Px + Txf;        // contributes to Fx = sum(-Px) + sum(Tx)
        fy = -Py + Tyf;
    }

    // Intra-wave (wave32) reduction of the 4 quantities.
    #pragma unroll
    for (int off = 16; off > 0; off >>= 1) {
        fx += __shfl_xor(fx, off, 32);
        fy += __shfl_xor(fy, off, 32);
        d3 += __shfl_xor(d3, off, 32);
        d4 += __shfl_xor(d4, off, 32);
    }
    if ((threadIdx.x & 31) == 0) {
        int w = threadIdx.x >> 5;            // wave 0..10
        s_part[w * 16 + 0] = fx;
        s_part[w * 16 + 1] = fy;
        s_part[w * 16 + 2] = d3;
        s_part[w * 16 + 3] = d4;
    }
    __syncthreads();

    // Cross-wave reduction on wave 0 via V_WMMA_F32_16X16X4_F32.
    // A(16x4): M = quantity, K = wave slice.  A layout (ISA 7.12.2):
    //   lanes 0-15 : VGPR0 = A[l,K0], VGPR1 = A[l,K1]
    //   lanes 16-31: VGPR0 = A[l,K2], VGPR1 = A[l,K3]
    // B = ones (layout-invariant) => D[m,n] += sum_k A[m,k].
    if (threadIdx.x < 32u) {
        const int lane = threadIdx.x;
        const int l    = lane & 15;
        v8f acc = {};
        v2f onesv;
        onesv[0] = 1.0f;
        onesv[1] = 1.0f;
        #pragma unroll
        for (int j = 0; j < 3; ++j) {        // waves 0..11 (rows >=11 are zero)
            const int w0 = 4 * j + ((lane < 16) ? 0 : 2);
            v2f a;
            a[0] = s_part[w0 * 16 + l];
            a[1] = s_part[(w0 + 1) * 16 + l];
            acc = __builtin_amdgcn_wmma_f32_16x16x4_f32(
                false, a, false, onesv, (short)0, acc, false, false);
        }
        if (lane == 0) {
            // D layout: VGPR v, lane 0 -> M = v, N = 0.
            float Fx  = acc[0];
            float Fy  = acc[1];
            float Sd3 = acc[2];
            float Sd4 = acc[3];

            float Ma = (Sd3 / (float)NODES) * 0.3f + 0.3f;
            float af = (Sd4 / (float)NODES) * ALPHA_F;
            // |af| ~ 0.006 rad: hardware transcendentals are plenty accurate
            // and avoid the libm Payne-Hanek range-reduction blob.
            float ca = __cosf(af);
            float sa = __sinf(af);
            float Fxn = Fx * ca + Fy * sa;
            float Fyn = Fy * ca - Fxn * sa;     // reference uses rotated Fx
            float vel = ACOUSTIC_F * Ma;
            float q   = 0.5f * 1.225f * vel * vel;
            out[2 * blockIdx.x + 0] = Fxn / q;  // Cd
            out[2 * blockIdx.x + 1] = Fyn / q;  // Cl
        }
    }
}

extern "C" void kernel_launch(void* const* d_in, const int* in_sizes, int n_in,
                              void* d_out, int out_size, void* d_ws, size_t ws_size,
                              hipStream_t stream) {
    // setup_inputs order: batch(i32), coords(f32), fields(f32), design(f32),
    //                     num_case(scalar), nodes_num(scalar)
    const float* coords = (const float*)d_in[1];
    const float* fields = (const float*)d_in[2];
    const float* design = (const float*)d_in[3];
    float*       out    = (float*)d_out;

    const int num_case = in_sizes[0] / CASE_NODES;   // batch has one entry/node

    dynamicor_case_kernel<<<num_case, BLOCK, 0, stream>>>(coords, fields, design, out);
}